// neural_rendering_crop_resize_31421980737549
// MI455X (gfx1250) — compile-verified
//
#include <hip/hip_runtime.h>
#include <hip/hip_bf16.h>

// ---------------------------------------------------------------------------
// neural rendering: vertex xform (WMMA f32 16x16x4) -> raster -> tex sample
// -> fused crop_and_resize.  Sizes fixed by reference: B=4,P=8,N=32,V=1002,
// F=1000, IMG=128, TGT=64, CH=8, TEX=1024.
// ---------------------------------------------------------------------------

#define NB 4
#define NP 8
#define NN 32          // N = B*P images
#define NV 1002
#define NF 1000
#define IMG 128
#define TGT 64
#define TEXD 1024
#define NCH 8
#define TRI_STRIDE 32  // floats per triangle setup record

typedef float v2f __attribute__((ext_vector_type(2)));
typedef float v8f __attribute__((ext_vector_type(8)));

// ---------------------------------------------------------------------------
// Kernel 1: M_n = cam_pose(n)^T @ PROJ  (one 4x4 per image, n = p*B + b)
// ---------------------------------------------------------------------------
__global__ __launch_bounds__(32) void build_mats_kernel(const float* __restrict__ poses,
                                                        float* __restrict__ mats) {
    int n = threadIdx.x;
    if (n >= NN) return;
    int b = n & (NB - 1), p = n >> 2;
    const float* P4 = poses + ((size_t)b * NP + p) * 16;   // poses[b][p] 4x4
    const float sgn[3] = {1.f, -1.f, -1.f};                // gl = diag(1,-1,-1)
    float cam[4][4];
    #pragma unroll
    for (int i = 0; i < 3; ++i) {
        #pragma unroll
        for (int j = 0; j < 4; ++j) cam[i][j] = sgn[i] * P4[i * 4 + j];
    }
    #pragma unroll
    for (int j = 0; j < 4; ++j) cam[3][j] = P4[12 + j];
    // PROJ (transposed build_projection with CAM_K, 128x128, nc=0.1, fc=10)
    const float q  = -(10.0f + 0.1f) / (10.0f - 0.1f);   // -1.0202020
    const float qn = -2.0f * 10.0f * 0.1f / (10.0f - 0.1f); // -0.2020202
    const float PR[4][4] = {{1.875f, 0.f, 0.f, 0.f},
                            {0.f, 1.875f, 0.f, 0.f},
                            {0.f, 0.f, q,   -1.f},
                            {0.f, 0.f, qn,   0.f}};
    // M[j][k] = sum_i cam[i][j] * PR[i][k]   (verts_clip = verts_h @ M)
    #pragma unroll
    for (int j = 0; j < 4; ++j)
        #pragma unroll
        for (int k = 0; k < 4; ++k) {
            float acc = 0.f;
            #pragma unroll
            for (int i = 0; i < 4; ++i) acc += cam[i][j] * PR[i][k];
            mats[n * 16 + j * 4 + k] = acc;
        }
}

// ---------------------------------------------------------------------------
// Kernel 2: WMMA vertex transform. One wave per 16-vertex tile.
// A (16x4 f32): VGPR0 = K0 (lanes0-15) / K2 (lanes16-31); VGPR1 = K1/K3.
// B (4x16 f32): VGPR0 = row K0 (lanes0-15) / K2 (lanes16-31); VGPR1 = K1/K3.
// D (16x16 f32): VGPR r = row r (lanes0-15) / row 8+r (lanes16-31), col=lane&15.
// Output vtx record (stride 8 f32): px,py,z,1/w_safe,w_ok,0,0,0
// ---------------------------------------------------------------------------
#define VTILES ((NV + 15) / 16)   // 63
__global__ __launch_bounds__(32) void vertex_xform_kernel(const float* __restrict__ vertices,
                                                          const float* __restrict__ mats,
                                                          float* __restrict__ vtx) {
    int gid  = blockIdx.x;
    int n    = gid / VTILES;
    int tile = gid % VTILES;
    int b    = n & (NB - 1);
    int lane = threadIdx.x;
    int half = lane >> 4;      // 0: K=0,1  1: K=2,3
    int l15  = lane & 15;

    v2f a; a[0] = 0.f; a[1] = 0.f;
    int row = tile * 16 + l15;
    if (row < NV) {
        const float* vp = vertices + ((size_t)b * NV + row) * 3;
        float h[4] = {vp[0], vp[1], vp[2], 1.0f};
        a[0] = h[half * 2 + 0];
        a[1] = h[half * 2 + 1];
    }
    v2f bm; bm[0] = 0.f; bm[1] = 0.f;
    if (l15 < 4) {
        const float* m = mats + n * 16;
        bm[0] = m[(half * 2 + 0) * 4 + l15];
        bm[1] = m[(half * 2 + 1) * 4 + l15];
    }
    v8f c = {};
    v8f d = __builtin_amdgcn_wmma_f32_16x16x4_f32(false, a, false, bm,
                                                  (short)0, c, false, false);
    __shared__ float clipLds[16][4];
    if (l15 < 4) {
        #pragma unroll
        for (int r = 0; r < 8; ++r) clipLds[r + half * 8][l15] = d[r];
    }
    __syncthreads();
    if (lane < 16) {
        int vrow = tile * 16 + lane;
        if (vrow < NV) {
            float X = clipLds[lane][0], Y = clipLds[lane][1];
            float Z = clipLds[lane][2], W = clipLds[lane][3];
            float wsafe = (fabsf(W) < 1e-8f) ? 1e-8f : W;
            float inv = 1.0f / wsafe;
            float px = (X * inv * 0.5f + 0.5f) * (float)IMG;
            float py = (1.0f - (Y * inv * 0.5f + 0.5f)) * (float)IMG;
            float z  = Z * inv;
            float wok = (W > 1e-8f) ? 1.0f : 0.0f;
            float* o = vtx + ((size_t)n * NV + vrow) * 8;
            float4 r0 = make_float4(px, py, z, inv);
            float4 r1 = make_float4(wok, 0.f, 0.f, 0.f);
            *(float4*)(o)     = r0;
            *(float4*)(o + 4) = r1;
        }
    }
}

// ---------------------------------------------------------------------------
// Kernel 3: per-triangle setup.  Record (stride 32 f32):
//  0..8  : sign-scaled edge coefs (A,B,C)x3  with e_i = A*gy + B*gx + C
//  9     : 1/|area|
// 10..12 : z0,z1,z2        13..15 : iw0,iw1,iw2
// 16..18 : iw*u             19..21 : iw*v
// 22     : valid (front && |area|>1e-9)
// 23..26 : bbox minx,miny,maxx,maxy
// ---------------------------------------------------------------------------
__global__ __launch_bounds__(256) void trisetup_kernel(const int* __restrict__ faces,
                                                       const float* __restrict__ uv_map,
                                                       const float* __restrict__ vtx,
                                                       float* __restrict__ tri) {
    int idx = blockIdx.x * 256 + threadIdx.x;
    if (idx >= NN * NF) return;
    int n = idx / NF, f = idx % NF;
    int b = n & (NB - 1);
    const int* fc = faces + ((size_t)b * NF + f) * 3;
    float X[3], Y[3], Z[3], IW[3], WK[3], U[3], Vt[3];
    #pragma unroll
    for (int k = 0; k < 3; ++k) {
        int vid = fc[k];
        const float* vr = vtx + ((size_t)n * NV + vid) * 8;
        X[k] = vr[0]; Y[k] = vr[1]; Z[k] = vr[2]; IW[k] = vr[3]; WK[k] = vr[4];
        const float* uvp = uv_map + ((size_t)b * NV + vid) * 2;
        U[k] = uvp[0]; Vt[k] = uvp[1];
    }
    float area = (X[1] - X[0]) * (Y[2] - Y[0]) - (Y[1] - Y[0]) * (X[2] - X[0]);
    float s = (area > 0.f) ? 1.f : ((area < 0.f) ? -1.f : 0.f);
    bool front = (WK[0] > 0.5f) && (WK[1] > 0.5f) && (WK[2] > 0.5f);
    bool valid = front && (fabsf(area) > 1e-9f);

    float* r = tri + (size_t)idx * TRI_STRIDE;
    // e0 = edge(v1,v2), e1 = edge(v2,v0), e2 = edge(v0,v1)
    // edge(a,b) = (bx-ax)*(gy-ay) - (by-ay)*(gx-ax)
    float A0 = X[2] - X[1], B0 = -(Y[2] - Y[1]), C0 = -A0 * Y[1] - B0 * X[1];
    float A1 = X[0] - X[2], B1 = -(Y[0] - Y[2]), C1 = -A1 * Y[2] - B1 * X[2];
    float A2 = X[1] - X[0], B2 = -(Y[1] - Y[0]), C2 = -A2 * Y[0] - B2 * X[0];
    r[0] = s * A0; r[1] = s * B0; r[2] = s * C0;
    r[3] = s * A1; r[4] = s * B1; r[5] = s * C1;
    r[6] = s * A2; r[7] = s * B2; r[8] = s * C2;
    r[9] = valid ? (1.0f / fabsf(area)) : 0.f;
    #pragma unroll
    for (int k = 0; k < 3; ++k) {
        r[10 + k] = Z[k];
        r[13 + k] = IW[k];
        r[16 + k] = IW[k] * U[k];
        r[19 + k] = IW[k] * Vt[k];
    }
    r[22] = valid ? 1.f : 0.f;
    r[23] = fminf(fminf(X[0], X[1]), X[2]);
    r[24] = fminf(fminf(Y[0], Y[1]), Y[2]);
    r[25] = fmaxf(fmaxf(X[0], X[1]), X[2]);
    r[26] = fmaxf(fmaxf(Y[0], Y[1]), Y[2]);
    r[27] = 0.f; r[28] = 0.f; r[29] = 0.f; r[30] = 0.f; r[31] = 0.f;
}

// ---------------------------------------------------------------------------
// Kernel 4: rasterizer.  Block = 16x16 pixel tile, 64 tiles/image.
// Triangles processed in reference scan order; LDS-staged 32 at a time.
// gbuf = (mask, u, v, depth) float4 per pixel.
// ---------------------------------------------------------------------------
__global__ __launch_bounds__(256) void raster_kernel(const float* __restrict__ tri,
                                                     float4* __restrict__ gbuf) {
    int blk = blockIdx.x;
    int n = blk >> 6;
    int t = blk & 63;
    int ty = t >> 3, tx = t & 7;
    int lx = threadIdx.x & 15, ly = threadIdx.x >> 4;
    int pxi = tx * 16 + lx, pyi = ty * 16 + ly;
    float gx = pxi + 0.5f, gy = pyi + 0.5f;
    float tMinX = tx * 16 + 0.5f, tMaxX = tx * 16 + 15.5f;
    float tMinY = ty * 16 + 0.5f, tMaxY = ty * 16 + 15.5f;

    __shared__ float sh[32 * TRI_STRIDE];
    const float* base = tri + (size_t)n * NF * TRI_STRIDE;

    float depth = __builtin_inff();
    float m = 0.f, u = 0.f, v = 0.f;

    for (int c0 = 0; c0 < NF; c0 += 32) {
        int cnt = min(32, NF - c0);
        __syncthreads();
        int fi = threadIdx.x * 4;          // 256 threads x float4 = 1024 floats
        if (fi < cnt * TRI_STRIDE) {
            float4 src = *(const float4*)(base + (size_t)c0 * TRI_STRIDE + fi);
            *(float4*)(&sh[fi]) = src;
        }
        __syncthreads();
        if (c0 + 32 < NF)  // gfx1250 global_prefetch_b8 for next chunk
            __builtin_prefetch(base + (size_t)(c0 + 32) * TRI_STRIDE, 0, 1);

        for (int j = 0; j < cnt; ++j) {
            const float* r = &sh[j * TRI_STRIDE];
            if (r[22] == 0.0f) continue;                       // invalid/backface
            if (r[25] < tMinX || r[23] > tMaxX ||
                r[26] < tMinY || r[24] > tMaxY) continue;      // tile bbox cull
            float e0 = fmaf(r[0], gy, fmaf(r[1], gx, r[2]));
            float e1 = fmaf(r[3], gy, fmaf(r[4], gx, r[5]));
            float e2 = fmaf(r[6], gy, fmaf(r[7], gx, r[8]));
            if (e0 < 0.f || e1 < 0.f || e2 < 0.f) continue;
            float invA = r[9];
            float b0 = e0 * invA, b1 = e1 * invA, b2 = e2 * invA;
            float zp = b0 * r[10] + b1 * r[11] + b2 * r[12];
            if (!(zp < depth) || zp < -1.f || zp > 1.f) continue;
            float pw = b0 * r[13] + b1 * r[14] + b2 * r[15];
            float pwc = (fabsf(pw) < 1e-12f) ? 1e-12f : pw;
            float ipw = 1.0f / pwc;
            depth = zp;
            m = pw * ipw;
            u = (b0 * r[16] + b1 * r[17] + b2 * r[18]) * ipw;
            v = (b0 * r[19] + b1 * r[20] + b2 * r[21]) * ipw;
        }
    }
    gbuf[((size_t)n * IMG + pyi) * IMG + pxi] = make_float4(m, u, v, depth);
}

// ---------------------------------------------------------------------------
// Kernel 5: bilinear texture sample (8 channels).  Texture set is 128 MB f32
// -> resident in the 192 MB L2; gather is L2-bandwidth bound, negligible.
// ---------------------------------------------------------------------------
__global__ __launch_bounds__(256) void texsample_kernel(const float4* __restrict__ gbuf,
                                                        const float* __restrict__ texture,
                                                        float* __restrict__ rend) {
    int idx = blockIdx.x * 256 + threadIdx.x;
    if (idx >= NN * IMG * IMG) return;
    int n = idx >> 14;
    int b = n & (NB - 1);
    float4 g = gbuf[idx];
    float uu = fminf(fmaxf(g.y, 0.f), 1.f);
    float vv = fminf(fmaxf(g.z, 0.f), 1.f);
    float iy = vv * (float)TEXD;    // uv[..., ::-1] -> row from v
    float ix = uu * (float)TEXD;
    float fly = floorf(iy), flx = floorf(ix);
    float fy = iy - fly, fx = ix - flx;
    int i = (int)fly, j = (int)flx;
    int i0 = min(max(i, 0), TEXD - 1), j0 = min(max(j, 0), TEXD - 1);
    int i1 = min(max(i + 1, 0), TEXD - 1), j1 = min(max(j + 1, 0), TEXD - 1);
    const float* tb = texture + (size_t)b * TEXD * TEXD * NCH;
    const float* tl = tb + ((size_t)i0 * TEXD + j0) * NCH;
    const float* tr = tb + ((size_t)i0 * TEXD + j1) * NCH;
    const float* bl = tb + ((size_t)i1 * TEXD + j0) * NCH;
    const float* br = tb + ((size_t)i1 * TEXD + j1) * NCH;
    float wtl = (1.f - fx) * (1.f - fy), wtr = fx * (1.f - fy);
    float wbl = (1.f - fx) * fy,         wbr = fx * fy;
    float* o = rend + (size_t)idx * NCH;
    #pragma unroll
    for (int h = 0; h < 2; ++h) {
        float4 a0 = *(const float4*)(tl + h * 4);
        float4 a1 = *(const float4*)(tr + h * 4);
        float4 a2 = *(const float4*)(bl + h * 4);
        float4 a3 = *(const float4*)(br + h * 4);
        float4 res;
        res.x = a0.x * wtl + a1.x * wtr + a2.x * wbl + a3.x * wbr;
        res.y = a0.y * wtl + a1.y * wtr + a2.y * wbl + a3.y * wbr;
        res.z = a0.z * wtl + a1.z * wtr + a2.z * wbl + a3.z * wbr;
        res.w = a0.w * wtl + a1.w * wtr + a2.w * wbl + a3.w * wbr;
        *(float4*)(o + h * 4) = res;
    }
}

// ---------------------------------------------------------------------------
// Kernel 6: fused crop_and_resize of 4 sources -> out (N,64,64,12):
//  ch0 = mask_new (cr of (u+v>0) > .5), ch1..8 = cr(rendered), ch9 = mask_old
//  (cr(gbuf mask) > .5), ch10..11 = cr(uv).
// ---------------------------------------------------------------------------
__global__ __launch_bounds__(256) void crop_kernel(const float4* __restrict__ gbuf,
                                                   const float* __restrict__ rend,
                                                   const float* __restrict__ bboxes,
                                                   float* __restrict__ out) {
    int idx = blockIdx.x * 256 + threadIdx.x;
    if (idx >= NN * TGT * TGT) return;
    int n = idx >> 12;
    int rem = idx & 4095;
    int oy = rem >> 6, ox = rem & 63;
    int b = n & (NB - 1), p = n >> 2;
    const float* bx = bboxes + ((size_t)b * NP + p) * 4;
    float y1 = bx[0], x1 = bx[1], y2 = bx[2], x2 = bx[3];
    const float Hm1 = (float)(IMG - 1);
    float ys = y1 * Hm1 + (float)oy * ((y2 - y1) * Hm1 / (float)(TGT - 1));
    float xs = x1 * Hm1 + (float)ox * ((x2 - x1) * Hm1 / (float)(TGT - 1));
    bool valid = (ys >= 0.f) && (ys <= Hm1) && (xs >= 0.f) && (xs <= Hm1);
    float yc = fminf(fmaxf(ys, 0.f), Hm1), xc = fminf(fmaxf(xs, 0.f), Hm1);
    float y0f = floorf(yc), x0f = floorf(xc);
    float fy = yc - y0f, fx = xc - x0f;
    int i0 = (int)y0f, j0 = (int)x0f;
    int i1 = min(i0 + 1, IMG - 1), j1 = min(j0 + 1, IMG - 1);
    float w00 = (1.f - fy) * (1.f - fx), w01 = (1.f - fy) * fx;
    float w10 = fy * (1.f - fx),         w11 = fy * fx;

    const float4* gb = gbuf + (size_t)n * IMG * IMG;
    float4 g00 = gb[i0 * IMG + j0], g01 = gb[i0 * IMG + j1];
    float4 g10 = gb[i1 * IMG + j0], g11 = gb[i1 * IMG + j1];

    float mask_s = w00 * g00.x + w01 * g01.x + w10 * g10.x + w11 * g11.x;
    float rm00 = (g00.y + g00.z > 0.f) ? 1.f : 0.f;
    float rm01 = (g01.y + g01.z > 0.f) ? 1.f : 0.f;
    float rm10 = (g10.y + g10.z > 0.f) ? 1.f : 0.f;
    float rm11 = (g11.y + g11.z > 0.f) ? 1.f : 0.f;
    float rm_s = w00 * rm00 + w01 * rm01 + w10 * rm10 + w11 * rm11;
    float u_s = w00 * g00.y + w01 * g01.y + w10 * g10.y + w11 * g11.y;
    float v_s = w00 * g00.z + w01 * g01.z + w10 * g10.z + w11 * g11.z;

    const float* rb = rend + (size_t)n * IMG * IMG * NCH;
    const float* r00 = rb + ((size_t)i0 * IMG + j0) * NCH;
    const float* r01 = rb + ((size_t)i0 * IMG + j1) * NCH;
    const float* r10 = rb + ((size_t)i1 * IMG + j0) * NCH;
    const float* r11 = rb + ((size_t)i1 * IMG + j1) * NCH;

    if (!valid) { mask_s = 0.f; rm_s = 0.f; u_s = 0.f; v_s = 0.f; }

    float* o = out + (size_t)idx * 12;
    o[0] = (rm_s > 0.5f) ? 1.f : 0.f;
    #pragma unroll
    for (int c = 0; c < NCH; ++c) {
        float rv = w00 * r00[c] + w01 * r01[c] + w10 * r10[c] + w11 * r11[c];
        o[1 + c] = valid ? rv : 0.f;
    }
    o[9]  = (mask_s > 0.5f) ? 1.f : 0.f;
    o[10] = u_s;
    o[11] = v_s;
}

// ---------------------------------------------------------------------------
// Launch
// ---------------------------------------------------------------------------
extern "C" void kernel_launch(void* const* d_in, const int* in_sizes, int n_in,
                              void* d_out, int out_size, void* d_ws, size_t ws_size,
                              hipStream_t stream) {
    const float* vertices = (const float*)d_in[0];  // (4,1002,3)
    const float* uv_map   = (const float*)d_in[1];  // (4,1002,2)
    const int*   faces    = (const int*)d_in[2];    // (4,1000,3)
    const float* texture  = (const float*)d_in[3];  // (4,1024,1024,8)
    const float* poses    = (const float*)d_in[4];  // (4,8,4,4)
    const float* bboxes   = (const float*)d_in[5];  // (4,8,4)
    float* out = (float*)d_out;                     // (8,4,64,64,12) flat
    float* w   = (float*)d_ws;

    // workspace layout (floats)
    float*  mats = w;                               // 512
    float*  vtx  = w + 512;                         // 32*1002*8 = 256512
    float*  tri  = w + 257024;                      // 32*1000*32 = 1024000
    float4* gbuf = (float4*)(w + 1281024);          // 32*128*128*4 = 2097152
    float*  rend = w + 3378176;                     // 32*128*128*8 = 4194304

    build_mats_kernel<<<1, 32, 0, stream>>>(poses, mats);
    vertex_xform_kernel<<<NN * VTILES, 32, 0, stream>>>(vertices, mats, vtx);
    trisetup_kernel<<<(NN * NF + 255) / 256, 256, 0, stream>>>(faces, uv_map, vtx, tri);
    raster_kernel<<<NN * 64, 256, 0, stream>>>(tri, gbuf);
    texsample_kernel<<<(NN * IMG * IMG) / 256, 256, 0, stream>>>(gbuf, texture, rend);
    crop_kernel<<<(NN * TGT * TGT) / 256, 256, 0, stream>>>(gbuf, rend, bboxes, out);
}